// HCAProtoNet_70179765617235
// MI455X (gfx1250) — compile-verified
//
#include <hip/hip_runtime.h>
#include <math.h>

typedef __attribute__((ext_vector_type(2))) float v2f;
typedef __attribute__((ext_vector_type(8))) float v8f;

#define DDIM 256
#define NPROTO 32   // 20 shared + 10 rare + 2 zero pad
#define NCLS 5

// ---------------------------------------------------------------------------
// Kernel 1: L2-normalize prototypes into padded 32x256 matrix P (workspace).
// rows 0..19 = shared_prototypes, rows 20..29 = rare_prototypes (flattened),
// rows 30..31 = zeros.
// grid = 32 blocks (one row each), block = 32 threads (one wave).
// ---------------------------------------------------------------------------
__global__ __launch_bounds__(32) void proto_norm_kernel(const float* __restrict__ sp,
                                                        const float* __restrict__ rp,
                                                        float* __restrict__ P) {
    const int r = blockIdx.x;       // 0..31
    const int lane = threadIdx.x;   // 0..31
    const float* src = nullptr;
    if (r < 20)      src = sp + (long)r * DDIM;
    else if (r < 30) src = rp + (long)(r - 20) * DDIM;

    float vals[8];
    float s = 0.0f;
    if (src) {
        #pragma unroll
        for (int i = 0; i < 8; ++i) {
            float v = src[i * 32 + lane];
            vals[i] = v;
            s += v * v;
        }
    } else {
        #pragma unroll
        for (int i = 0; i < 8; ++i) vals[i] = 0.0f;
    }
    #pragma unroll
    for (int off = 16; off > 0; off >>= 1) s += __shfl_xor(s, off, 32);
    const float inv = 1.0f / fmaxf(sqrtf(s), 1e-12f);
    #pragma unroll
    for (int i = 0; i < 8; ++i) P[(long)r * DDIM + i * 32 + lane] = vals[i] * inv;
}

// ---------------------------------------------------------------------------
// Kernel 2: main fused kernel. 1 block = 16 rows of x, 128 threads = 4 waves.
// ---------------------------------------------------------------------------
__global__ __launch_bounds__(128) void hca_main_kernel(const float* __restrict__ x,
                                                       const float* __restrict__ W,       // 20x5
                                                       const float* __restrict__ rarity,  // 5
                                                       const float* __restrict__ P,       // 32x256 normalized
                                                       float* __restrict__ out_logits,    // Bx5
                                                       float* __restrict__ out_feat,      // Bx256
                                                       float* __restrict__ out_ss,        // Bx20
                                                       float* __restrict__ out_lr) {      // Bx5
    __shared__ float xs[16 * DDIM];          // 16 KB : normalized x tile
    __shared__ float ps[NPROTO * DDIM];      // 32 KB : normalized prototypes
    __shared__ float simw[4 * 16 * NPROTO];  //  8 KB : per-wave split-K partials
    __shared__ float fin[16 * NPROTO];       //  2 KB : reduced sims
    __shared__ float Ws[20 * NCLS];
    __shared__ float rar[NCLS];
    __shared__ float nrm2[16];               // sumsq -> inv-norm per row

    const int tid  = threadIdx.x;
    const int wave = tid >> 5;
    const int lane = tid & 31;
    const long rowBase = (long)blockIdx.x * 16;

    if (tid < 16) nrm2[tid] = 0.0f;
    if (tid < 20 * NCLS) Ws[tid] = W[tid];
    if (tid < NCLS) rar[tid] = rarity[tid];
    __syncthreads();

    // ---- Phase 1: stream x tile in (coalesced), echo features, row sumsq ----
    #pragma unroll 4
    for (int r = 0; r < 16; ++r) {
        const long g = (rowBase + r) * DDIM;
        float v0 = x[g + tid];
        float v1 = x[g + 128 + tid];
        xs[r * DDIM + tid]       = v0;
        xs[r * DDIM + 128 + tid] = v1;
        out_feat[g + tid]        = v0;
        out_feat[g + 128 + tid]  = v1;
        float s = v0 * v0 + v1 * v1;
        #pragma unroll
        for (int off = 16; off > 0; off >>= 1) s += __shfl_xor(s, off, 32);
        if (lane == 0) atomicAdd(&nrm2[r], s);  // ds_add_f32, 4 adds per row
    }
    // ---- prototypes into LDS (hits L2 after first block) ----
    for (int i = tid; i < NPROTO * DDIM; i += 128) ps[i] = P[i];
    __syncthreads();

    if (tid < 16) nrm2[tid] = 1.0f / fmaxf(sqrtf(nrm2[tid]), 1e-12f);
    __syncthreads();

    // ---- scale tile in place: xs = fn ----
    #pragma unroll 4
    for (int r = 0; r < 16; ++r) {
        const float sc = nrm2[r];
        xs[r * DDIM + tid]       *= sc;
        xs[r * DDIM + 128 + tid] *= sc;
    }
    __syncthreads();

    // ---- Phase 2: WMMA, f32 16x16x4. Wave w owns K slice [w*64, w*64+64). ----
    // A 16x4 f32 layout: lane l -> M = l&15, K-half = l>>4; VGPR pair holds
    // {A[m][2h], A[m][2h+1]}. B 4x16 mirrors with N = l&15.
    const int mn    = lane & 15;
    const int khalf = lane >> 4;
    const int kb0   = wave * 64 + 2 * khalf;

    v8f c0 = {};  // sims vs prototypes 0..15
    v8f c1 = {};  // sims vs prototypes 16..31
    #pragma unroll
    for (int t = 0; t < 16; ++t) {
        const int kb = kb0 + t * 4;
        v2f a  = *(const v2f*)&xs[mn * DDIM + kb];
        v2f b0 = *(const v2f*)&ps[mn * DDIM + kb];
        v2f b1 = *(const v2f*)&ps[(16 + mn) * DDIM + kb];
        c0 = __builtin_amdgcn_wmma_f32_16x16x4_f32(false, a, false, b0, (short)0, c0,
                                                   false, false);
        c1 = __builtin_amdgcn_wmma_f32_16x16x4_f32(false, a, false, b1, (short)0, c1,
                                                   false, false);
    }
    // C/D layout: VGPR r -> M = r + 8*khalf, N = lane&15
    #pragma unroll
    for (int r = 0; r < 8; ++r) {
        const int m = r + 8 * khalf;
        simw[wave * 512 + m * NPROTO + mn]      = c0[r];
        simw[wave * 512 + m * NPROTO + 16 + mn] = c1[r];
    }
    __syncthreads();

    // ---- split-K reduction ----
    for (int i = tid; i < 16 * NPROTO; i += 128)
        fin[i] = simw[i] + simw[512 + i] + simw[1024 + i] + simw[1536 + i];
    __syncthreads();

    // ---- Phase 3: finalize, one lane per row ----
    if (tid < 16) {
        const int m = tid;
        const long grow = rowBase + m;
        const float* sv = &fin[m * NPROTO];

        #pragma unroll
        for (int k = 0; k < 20; ++k) out_ss[grow * 20 + k] = sv[k];

        float lg[NCLS];
        #pragma unroll
        for (int c = 0; c < NCLS; ++c) {
            float acc = 0.0f;
            #pragma unroll
            for (int k = 0; k < 20; ++k) acc += sv[k] * Ws[k * NCLS + c];
            lg[c] = acc;
        }
        // softmax(logits/1.5), entropy, uncertainty
        float z[NCLS], zmax = -3.4e38f;
        #pragma unroll
        for (int c = 0; c < NCLS; ++c) {
            z[c] = lg[c] * (1.0f / 1.5f);
            zmax = fmaxf(zmax, z[c]);
        }
        float p[NCLS], se = 0.0f;
        #pragma unroll
        for (int c = 0; c < NCLS; ++c) { p[c] = expf(z[c] - zmax); se += p[c]; }
        const float invse = 1.0f / se;
        float H = 0.0f;
        #pragma unroll
        for (int c = 0; c < NCLS; ++c) {
            float pc = p[c] * invse;
            H -= pc * logf(pc + 1e-8f);
        }
        const float unc = H * (1.0f / 1.6094379124341003f);  // / log(5)

        float m0 = sv[20], m1 = sv[25];
        #pragma unroll
        for (int k = 1; k < 5; ++k) {
            m0 = fmaxf(m0, sv[20 + k]);
            m1 = fmaxf(m1, sv[25 + k]);
        }
        const float lr0 = rar[0] * unc * m0;
        const float lr1 = rar[1] * unc * m1;

        out_lr[grow * NCLS + 0] = lr0;
        out_lr[grow * NCLS + 1] = lr1;
        out_lr[grow * NCLS + 2] = 0.0f;
        out_lr[grow * NCLS + 3] = 0.0f;
        out_lr[grow * NCLS + 4] = 0.0f;

        out_logits[grow * NCLS + 0] = lg[0] + lr0;
        out_logits[grow * NCLS + 1] = lg[1] + lr1;
        out_logits[grow * NCLS + 2] = lg[2];
        out_logits[grow * NCLS + 3] = lg[3];
        out_logits[grow * NCLS + 4] = lg[4];
    }
}

extern "C" void kernel_launch(void* const* d_in, const int* in_sizes, int n_in,
                              void* d_out, int out_size, void* d_ws, size_t ws_size,
                              hipStream_t stream) {
    const float* x  = (const float*)d_in[0];  // (B, 256)
    const float* sp = (const float*)d_in[1];  // (20, 256)
    const float* W  = (const float*)d_in[2];  // (20, 5)
    const float* rp = (const float*)d_in[3];  // (2, 5, 256)
    const float* rf = (const float*)d_in[4];  // (5,)

    const long B = (long)in_sizes[0] / DDIM;

    float* P = (float*)d_ws;  // 32x256 normalized prototypes (32 KB)

    float* out        = (float*)d_out;
    float* out_logits = out;                     // B*5
    float* out_feat   = out + B * 5;             // B*256
    float* out_ss     = out + B * (5 + 256);     // B*20
    float* out_lr     = out + B * (5 + 256 + 20);// B*5

    proto_norm_kernel<<<dim3(NPROTO), dim3(32), 0, stream>>>(sp, rp, P);
    hca_main_kernel<<<dim3((unsigned)(B / 16)), dim3(128), 0, stream>>>(
        x, W, rf, P, out_logits, out_feat, out_ss, out_lr);
}